// EMD_loss_36197984371226
// MI455X (gfx1250) — compile-verified
//
#include <hip/hip_runtime.h>
#include <math.h>

typedef __attribute__((ext_vector_type(2))) float v2f;
typedef __attribute__((ext_vector_type(8))) float v8f;

#define N_PTS 4096
#define TILE 16
#define NTILES (N_PTS / TILE)          // 256 tiles per dimension
#define WAVES_PER_BLOCK 8
#define BLOCK_THREADS (WAVES_PER_BLOCK * 32)

// ---------------------------------------------------------------------------
// Pad 3-D points to float4 (w = 0 so the K=4 WMMA pad contributes nothing)
// and compute half squared norms.
// ---------------------------------------------------------------------------
__global__ __launch_bounds__(256)
void prep_kernel(const float* __restrict__ x, const float* __restrict__ y,
                 float* __restrict__ xp, float* __restrict__ sx,
                 float* __restrict__ yp, float* __restrict__ sy)
{
    int i = blockIdx.x * blockDim.x + threadIdx.x;
    if (i < N_PTS) {
        float a = x[3 * i], b = x[3 * i + 1], c = x[3 * i + 2];
        xp[4 * i + 0] = a; xp[4 * i + 1] = b; xp[4 * i + 2] = c; xp[4 * i + 3] = 0.f;
        sx[i] = 0.5f * (a * a + b * b + c * c);
    } else if (i < 2 * N_PTS) {
        int j = i - N_PTS;
        float a = y[3 * j], b = y[3 * j + 1], c = y[3 * j + 2];
        yp[4 * j + 0] = a; yp[4 * j + 1] = b; yp[4 * j + 2] = c; yp[4 * j + 3] = 0.f;
        sy[j] = 0.5f * (a * a + b * b + c * c);
    }
}

__global__ __launch_bounds__(256)
void zero_fg_kernel(float* __restrict__ f, float* __restrict__ g)
{
    int i = blockIdx.x * blockDim.x + threadIdx.x;
    if (i < N_PTS) { f[i] = 0.f; g[i] = 0.f; }
}

// ---------------------------------------------------------------------------
// One Sinkhorn half-step:  f_out[i] = -eps * logsumexp_j( lb + (g_j - C_ij)/eps )
// with C_ij = rowSq[i] + colSq[j] - row_i . col_j, recomputed on the fly.
//
// WMMA f32 16x16x4 computes the whole logit tile:
//   A = row-points * (1/eps)        (16x4, K padded with 0)
//   B = col-points                  (4x16)
//   C = (g_n - colSq_n)/eps + (lb - rowSq_m/eps)
//   D = A x B + C  ==  lb + (g_j - C_ij)/eps      (the logit, directly)
// Each wave owns a strided set of 16-wide column tiles and keeps an online
// logsumexp state for its 8 C/D rows; cross-lane merge via shfl_xor (wave32,
// rows live entirely inside a 16-lane half-group), cross-wave merge via LDS.
// ---------------------------------------------------------------------------
__global__ __launch_bounds__(BLOCK_THREADS)
void sinkhorn_half_kernel(const float* __restrict__ rowP, const float* __restrict__ rowSq,
                          const float* __restrict__ colP, const float* __restrict__ colSq,
                          const float* __restrict__ g_in, float* __restrict__ f_out,
                          float inv_eps, float eps, float lb)
{
    const int tid  = threadIdx.x;
    const int lane = tid & 31;
    const int wave = tid >> 5;
    const int nlo  = lane & 15;   // row (A) / col (B,C,D) index within tile
    const int hi   = lane >> 4;   // half-group: K in {0,1} vs {2,3}; rows +8 for C/D
    const int rowBase = blockIdx.x * TILE;

    // A operand (constant across the column loop), pre-scaled by 1/eps.
    const int k0 = hi * 2;
    v2f a;
    a.x = rowP[(rowBase + nlo) * 4 + k0 + 0] * inv_eps;
    a.y = rowP[(rowBase + nlo) * 4 + k0 + 1] * inv_eps;

    // Per-lane row bias for the 8 D rows this lane holds: lb - rowSq_m / eps.
    float bm[8];
#pragma unroll
    for (int v = 0; v < 8; ++v)
        bm[v] = lb - rowSq[rowBase + v + 8 * hi] * inv_eps;

    // Online logsumexp state per held row.
    float mx[8], sm[8];
#pragma unroll
    for (int v = 0; v < 8; ++v) { mx[v] = -__builtin_inff(); sm[v] = 0.f; }

    for (int t = wave; t < NTILES; t += WAVES_PER_BLOCK) {
        const int n = t * TILE + nlo;
        v2f b;
        b.x = colP[n * 4 + k0 + 0];
        b.y = colP[n * 4 + k0 + 1];
        const float cn = (g_in[n] - colSq[n]) * inv_eps;
        v8f cacc;
#pragma unroll
        for (int v = 0; v < 8; ++v) cacc[v] = cn + bm[v];

        // D = (row . col)/eps + bias  ==  the Sinkhorn logit for this tile.
        v8f d = __builtin_amdgcn_wmma_f32_16x16x4_f32(
            /*neg_a=*/false, a, /*neg_b=*/false, b,
            /*c_mod=*/(short)0, cacc, /*reuse_a=*/false, /*reuse_b=*/false);

#pragma unroll
        for (int v = 0; v < 8; ++v) {
            float val = d[v];
            float nm  = fmaxf(mx[v], val);
            sm[v] = sm[v] * __expf(mx[v] - nm) + __expf(val - nm);
            mx[v] = nm;
        }
    }

    // Reduce the 16 columns held across this half-group's lanes.
#pragma unroll
    for (int mask = 1; mask < 16; mask <<= 1) {
#pragma unroll
        for (int v = 0; v < 8; ++v) {
            float om = __shfl_xor(mx[v], mask, 32);
            float os = __shfl_xor(sm[v], mask, 32);
            float nm = fmaxf(mx[v], om);
            sm[v] = sm[v] * __expf(mx[v] - nm) + os * __expf(om - nm);
            mx[v] = nm;
        }
    }

    __shared__ float lmx[TILE][WAVES_PER_BLOCK];
    __shared__ float lsm[TILE][WAVES_PER_BLOCK];
    if (nlo == 0) {
#pragma unroll
        for (int v = 0; v < 8; ++v) {
            lmx[v + 8 * hi][wave] = mx[v];
            lsm[v + 8 * hi][wave] = sm[v];
        }
    }
    __syncthreads();

    if (tid < TILE) {
        float m = lmx[tid][0], s = lsm[tid][0];
#pragma unroll
        for (int w = 1; w < WAVES_PER_BLOCK; ++w) {
            float om = lmx[tid][w], os = lsm[tid][w];
            float nm = fmaxf(m, om);
            s = s * __expf(m - nm) + os * __expf(om - nm);
            m = nm;
        }
        f_out[rowBase + tid] = -eps * (m + __logf(s));
    }
}

// ---------------------------------------------------------------------------
// out[0] (+)= weight * (mean(f) + mean(g))
// ---------------------------------------------------------------------------
__global__ __launch_bounds__(256)
void reduce_fg_kernel(const float* __restrict__ f, const float* __restrict__ g,
                      float* __restrict__ out, float weight, int accumulate)
{
    __shared__ float buf[256];
    float s = 0.f;
    for (int i = threadIdx.x; i < N_PTS; i += 256) s += f[i] + g[i];
    buf[threadIdx.x] = s;
    __syncthreads();
    for (int off = 128; off > 0; off >>= 1) {
        if (threadIdx.x < off) buf[threadIdx.x] += buf[threadIdx.x + off];
        __syncthreads();
    }
    if (threadIdx.x == 0) {
        float r = weight * (buf[0] / (float)N_PTS);
        if (accumulate) out[0] += r; else out[0] = r;
    }
}

extern "C" void kernel_launch(void* const* d_in, const int* in_sizes, int n_in,
                              void* d_out, int out_size, void* d_ws, size_t ws_size,
                              hipStream_t stream)
{
    (void)in_sizes; (void)n_in; (void)out_size; (void)ws_size;

    const float* x = (const float*)d_in[0];
    const float* y = (const float*)d_in[1];
    float* out = (float*)d_out;

    float* xp = (float*)d_ws;          // 4096 * 4
    float* yp = xp + N_PTS * 4;        // 4096 * 4
    float* sx = yp + N_PTS * 4;        // 4096
    float* sy = sx + N_PTS;            // 4096
    float* f  = sy + N_PTS;            // 4096
    float* g  = f  + N_PTS;            // 4096   (192 KB total)

    prep_kernel<<<(2 * N_PTS) / 256, 256, 0, stream>>>(x, y, xp, sx, yp, sy);

    // geomloss schedule: eps = 4 * 0.25^k while > 0.0025, then 5 x 0.0025
    const float eps_sched[11] = {4.f, 1.f, 0.25f, 0.0625f, 0.015625f, 0.00390625f,
                                 0.0025f, 0.0025f, 0.0025f, 0.0025f, 0.0025f};
    const float lb = -logf((float)N_PTS);   // la == lb (uniform weights, N == M)

    // Problems: OT(x,y) - 0.5*OT(x,x) - 0.5*OT(y,y)
    const float* RP[3] = {xp, xp, yp};
    const float* RS[3] = {sx, sx, sy};
    const float* CP[3] = {yp, xp, yp};
    const float* CS[3] = {sy, sx, sy};
    const float  W[3]  = {1.f, -0.5f, -0.5f};

    for (int p = 0; p < 3; ++p) {
        zero_fg_kernel<<<N_PTS / 256, 256, 0, stream>>>(f, g);
        for (int it = 0; it < 11; ++it) {
            const float eps = eps_sched[it];
            const float ie  = 1.f / eps;
            // f-update: reduce over columns of C (g_in = g)
            sinkhorn_half_kernel<<<NTILES, BLOCK_THREADS, 0, stream>>>(
                RP[p], RS[p], CP[p], CS[p], g, f, ie, eps, lb);
            // g-update: reduce over rows of C == columns of C^T (g_in = new f)
            sinkhorn_half_kernel<<<NTILES, BLOCK_THREADS, 0, stream>>>(
                CP[p], CS[p], RP[p], RS[p], f, g, ie, eps, lb);
        }
        reduce_fg_kernel<<<1, 256, 0, stream>>>(f, g, out, W[p], p > 0);
    }
}